// SUSAGEBin_35485019799946
// MI455X (gfx1250) — compile-verified
//
#include <hip/hip_runtime.h>
#include <hip/hip_bf16.h>

#define N_NODES 100000
#define N_EDGES 1000000
#define DIM 64

typedef __attribute__((ext_vector_type(2))) float v2f;
typedef __attribute__((ext_vector_type(8))) float v8f;

// ---------------------------------------------------------------- zero fill
__global__ void zero_kernel(float* __restrict__ p, long long n) {
    long long i = (long long)blockIdx.x * blockDim.x + threadIdx.x;
    long long stride = (long long)gridDim.x * blockDim.x;
    for (; i < n; i += stride) p[i] = 0.0f;
}

// ------------------------------------------- edge scatter (mean numerator)
// One thread per (edge, 4-dim quad): 16 threads per edge. Optionally also
// accumulates destination degree (layer 0 only; degree is reused in layer 1).
__global__ void scatter_kernel(const float* __restrict__ feat,
                               const int* __restrict__ eidx,
                               float* __restrict__ sum,
                               float* __restrict__ deg,
                               int add_deg) {
    long long tid = (long long)blockIdx.x * blockDim.x + threadIdx.x;
    if (tid >= 16LL * N_EDGES) return;
    int e = (int)(tid >> 4);
    int q = (int)(tid & 15);
    int src = eidx[e];
    int dst = eidx[N_EDGES + e];
    const float4* f = (const float4*)(feat + (long long)src * DIM) + q;
    float4 v = *f;
    float* s = sum + (long long)dst * DIM + q * 4;
    atomicAdd(s + 0, v.x);
    atomicAdd(s + 1, v.y);
    atomicAdd(s + 2, v.z);
    atomicAdd(s + 3, v.w);
    if (add_deg && q == 0) atomicAdd(deg + dst, 1.0f);
}

// --------------------------------------------- layer 0: WMMA fp32 SAGEConv
// One wave per 16-node tile. h1 = relu(mean_nb @ Wl0^T + bl0 + x @ Wr0^T).
// Uses V_WMMA_F32_16X16X4_F32 (exact fp32), 128 wmma ops per wave.
__global__ void sage_layer0_wmma(const float* __restrict__ x,
                                 const float* __restrict__ sum0,
                                 const float* __restrict__ deg,
                                 const float* __restrict__ Wl,
                                 const float* __restrict__ bl,
                                 const float* __restrict__ Wr,
                                 float* __restrict__ h1) {
    int wave = (int)(((long long)blockIdx.x * blockDim.x + threadIdx.x) >> 5);
    int lane = threadIdx.x & 31;
    const int ntiles = N_NODES / 16;          // 6250, exact
    if (wave >= ntiles) return;               // wave-uniform: EXEC stays all-1s
    int node0 = wave * 16;

    // A-matrix layout (16x4 f32): lane L -> row L&15, k-half (L>>4)*2
    int r  = lane & 15;
    int kh = (lane >> 4) << 1;
    const float* srow = sum0 + (size_t)(node0 + r) * DIM;
    const float* xrow = x    + (size_t)(node0 + r) * DIM;
    float dinv = 1.0f / fmaxf(deg[node0 + r], 1.0f);

    // Per-lane A fragments for all 16 k-steps (fully register resident).
    float am[32], ar[32];
#pragma unroll
    for (int kk = 0; kk < 16; ++kk) {
        int k = kk * 4 + kh;
        am[2 * kk]     = srow[k]     * dinv;
        am[2 * kk + 1] = srow[k + 1] * dinv;
        ar[2 * kk]     = xrow[k];
        ar[2 * kk + 1] = xrow[k + 1];
    }

    // C/D layout: lane L, vgpr v -> M = v + 8*(L>>4), N = L&15
    int n0 = lane & 15;
    int mb = (lane >> 4) << 3;

    for (int jt = 0; jt < 4; ++jt) {          // 4 tiles of 16 output columns
        v8f acc = {};
        // GEMM 1: mean_nb @ Wl^T  (B[k][n] = Wl[n][k], contiguous pair load)
#pragma unroll
        for (int kk = 0; kk < 16; ++kk) {
            int k = kk * 4 + kh;
            v2f a; a.x = am[2 * kk]; a.y = am[2 * kk + 1];
            v2f b = *(const v2f*)(Wl + (size_t)(jt * 16 + n0) * DIM + k);
            acc = __builtin_amdgcn_wmma_f32_16x16x4_f32(
                false, a, false, b, (short)0, acc, false, false);
        }
        // GEMM 2: x_root @ Wr^T, accumulated into the same C
#pragma unroll
        for (int kk = 0; kk < 16; ++kk) {
            int k = kk * 4 + kh;
            v2f a; a.x = ar[2 * kk]; a.y = ar[2 * kk + 1];
            v2f b = *(const v2f*)(Wr + (size_t)(jt * 16 + n0) * DIM + k);
            acc = __builtin_amdgcn_wmma_f32_16x16x4_f32(
                false, a, false, b, (short)0, acc, false, false);
        }
        float bias = bl[jt * 16 + n0];
#pragma unroll
        for (int v = 0; v < 8; ++v) {
            float val = acc[v] + bias;
            val = fmaxf(val, 0.0f);           // ReLU (dropout = identity in eval)
            h1[(size_t)(node0 + mb + v) * DIM + jt * 16 + n0] = val;
        }
    }
}

// --------------------------------- layer 1: per-node dot (out dim = 1) ----
__global__ void sage_layer1(const float* __restrict__ h1,
                            const float* __restrict__ sum1,
                            const float* __restrict__ deg,
                            const float* __restrict__ Wl1,
                            const float* __restrict__ bl1,
                            const float* __restrict__ Wr1,
                            float* __restrict__ out) {
    int n = blockIdx.x * blockDim.x + threadIdx.x;
    if (n >= N_NODES) return;
    float dinv = 1.0f / fmaxf(deg[n], 1.0f);
    const float4* s  = (const float4*)(sum1 + (size_t)n * DIM);
    const float4* h  = (const float4*)(h1   + (size_t)n * DIM);
    const float4* wl = (const float4*)Wl1;
    const float4* wr = (const float4*)Wr1;
    float accl = 0.0f, accr = 0.0f;
#pragma unroll
    for (int i = 0; i < 16; ++i) {
        float4 sv = s[i], hv = h[i], wlv = wl[i], wrv = wr[i];
        accl += sv.x * wlv.x + sv.y * wlv.y + sv.z * wlv.z + sv.w * wlv.w;
        accr += hv.x * wrv.x + hv.y * wrv.y + hv.z * wrv.z + hv.w * wrv.w;
    }
    float v = accl * dinv + bl1[0] + accr;
    out[n] = v;
    out[N_NODES + n] = 1.0f / (1.0f + __expf(-v));   // sigmoid
}

// --------------------------------------------------------------------------
extern "C" void kernel_launch(void* const* d_in, const int* in_sizes, int n_in,
                              void* d_out, int out_size, void* d_ws, size_t ws_size,
                              hipStream_t stream) {
    (void)in_sizes; (void)n_in; (void)out_size; (void)ws_size;

    const float* x   = (const float*)d_in[0];
    const int*   ei  = (const int*)d_in[1];
    const float* Wl0 = (const float*)d_in[2];
    const float* bl0 = (const float*)d_in[3];
    const float* Wr0 = (const float*)d_in[4];
    const float* Wl1 = (const float*)d_in[5];
    const float* bl1 = (const float*)d_in[6];
    const float* Wr1 = (const float*)d_in[7];
    float* out = (float*)d_out;

    // Workspace layout (all fp32): deg | sum0 | h1 | sum1  (~77 MB)
    float* deg  = (float*)d_ws;
    float* sum0 = deg  + N_NODES;
    float* h1   = sum0 + (size_t)N_NODES * DIM;
    float* sum1 = h1   + (size_t)N_NODES * DIM;

    // 1) zero deg+sum0 (contiguous) and sum1 — must re-init every call.
    long long z1 = (long long)N_NODES + (long long)N_NODES * DIM;
    long long z2 = (long long)N_NODES * DIM;
    zero_kernel<<<2048, 256, 0, stream>>>(deg, z1);
    zero_kernel<<<2048, 256, 0, stream>>>(sum1, z2);

    // 2) layer-0 scatter (+degree): 16 threads/edge
    const long long st = 16LL * N_EDGES;
    int sblocks = (int)((st + 255) / 256);                 // 62500
    scatter_kernel<<<sblocks, 256, 0, stream>>>(x, ei, sum0, deg, 1);

    // 3) layer-0 WMMA SAGEConv -> h1
    int waves  = N_NODES / 16;                             // 6250
    int tblocks = (waves * 32 + 255) / 256;
    sage_layer0_wmma<<<tblocks, 256, 0, stream>>>(x, sum0, deg, Wl0, bl0, Wr0, h1);

    // 4) layer-1 scatter of h1 (degree reused)
    scatter_kernel<<<sblocks, 256, 0, stream>>>(h1, ei, sum1, deg, 0);

    // 5) layer-1 dot + sigmoid -> d_out[0:N] = h, d_out[N:2N] = sigmoid(h)
    sage_layer1<<<(N_NODES + 255) / 256, 256, 0, stream>>>(h1, sum1, deg,
                                                           Wl1, bl1, Wr1, out);
}